// Gconv_last_33809982554693
// MI455X (gfx1250) — compile-verified
//
#include <hip/hip_runtime.h>

typedef __attribute__((ext_vector_type(2))) float v2f;
typedef __attribute__((ext_vector_type(8))) float v8f;

#define BS   2
#define NN   4096
#define JJ   3
#define F_IN 32
#define KDIM (NN * JJ)                 // 12288 : reduction length per output row
#define WAVES_PER_WG 8
#define THREADS (WAVES_PER_WG * 32)
#define KSLICE (KDIM / WAVES_PER_WG)   // 1536 per wave

// ---------------------------------------------------------------------------
// Kernel 1: fold fc_w into x:  y[b,m,j] = sum_f x[b,m,f] * fc_w[j*F_IN + f]
// ---------------------------------------------------------------------------
__global__ void gconv_fold_fc(const float* __restrict__ x,
                              const float* __restrict__ fc_w,
                              float* __restrict__ y) {
    int idx = blockIdx.x * blockDim.x + threadIdx.x;   // (b*NN + m)*JJ + j
    if (idx >= BS * NN * JJ) return;
    int j  = idx % JJ;
    int bm = idx / JJ;
    const float* xp = x + (size_t)bm * F_IN;
    const float* wp = fc_w + j * F_IN;
    float s = 0.f;
#pragma unroll
    for (int f = 0; f < F_IN; ++f) s += xp[f] * wp[f];
    y[idx] = s;
}

// ---------------------------------------------------------------------------
// Kernel 2: out[b,n] = fc_b + sum_k W[b,n,k] * y[b,k]   (k = m*JJ + j)
// One 16-row tile per workgroup; 8 waves split k; V_WMMA_F32_16X16X4_F32 does
// the 16-row x 4-k rank-update with y broadcast across all 16 B-columns.
// Uniform SGPR base + 32-bit lane offsets -> saddr-form global_load_b64 with
// immediate offsets inside the unrolled body.
// ---------------------------------------------------------------------------
__global__ __launch_bounds__(THREADS)
void gconv_gemv_wmma(const float* __restrict__ W,
                     const float* __restrict__ y,
                     const float* __restrict__ fc_b,
                     float* __restrict__ out) {
    __shared__ float partial[WAVES_PER_WG * 16];

    const int tile = blockIdx.x;               // 0 .. BS*NN/16 - 1
    const int b    = tile / (NN / 16);
    const int n0   = (tile % (NN / 16)) * 16;

    const int wave = (int)(threadIdx.x >> 5);
    const int lane = (int)(threadIdx.x & 31);
    const int row  = lane & 15;                // A-matrix M index for this lane
    const int half = lane >> 4;                // 0: holds k,k+1   1: holds k+2,k+3

    // Uniform (per-block) base pointers; only 32-bit element offsets diverge.
    const float* __restrict__ Wb = W + (size_t)(b * NN + n0) * KDIM;
    const float* __restrict__ yb = y + (size_t)b * KDIM;

    const unsigned wlane = (unsigned)(row * KDIM + (half << 1)); // < 768K floats
    const unsigned ylane = (unsigned)(half << 1);

    const int kbeg = wave * KSLICE;
    const int kend = kbeg + KSLICE;

    v8f acc0 = {}, acc1 = {}, acc2 = {}, acc3 = {};
#pragma unroll 2
    for (int k = kbeg; k < kend; k += 16) {
        const unsigned wo = wlane + (unsigned)k;
        const unsigned yo = ylane + (unsigned)k;
        v2f a0 = *(const v2f*)(Wb + wo);
        v2f b0 = *(const v2f*)(yb + yo);
        acc0 = __builtin_amdgcn_wmma_f32_16x16x4_f32(
                   false, a0, false, b0, (short)0, acc0, false, false);
        v2f a1 = *(const v2f*)(Wb + wo + 4);
        v2f b1 = *(const v2f*)(yb + yo + 4);
        acc1 = __builtin_amdgcn_wmma_f32_16x16x4_f32(
                   false, a1, false, b1, (short)0, acc1, false, false);
        v2f a2 = *(const v2f*)(Wb + wo + 8);
        v2f b2 = *(const v2f*)(yb + yo + 8);
        acc2 = __builtin_amdgcn_wmma_f32_16x16x4_f32(
                   false, a2, false, b2, (short)0, acc2, false, false);
        v2f a3 = *(const v2f*)(Wb + wo + 12);
        v2f b3 = *(const v2f*)(yb + yo + 12);
        acc3 = __builtin_amdgcn_wmma_f32_16x16x4_f32(
                   false, a3, false, b3, (short)0, acc3, false, false);
    }
    v8f acc = (acc0 + acc1) + (acc2 + acc3);

    // Every B-column was the broadcast y, so all 16 columns of D are equal.
    // Lane 0 holds rows 0..7 in acc[0..7]; lane 16 holds rows 8..15.
    if (lane == 0) {
#pragma unroll
        for (int r = 0; r < 8; ++r) partial[wave * 16 + r] = acc[r];
    } else if (lane == 16) {
#pragma unroll
        for (int r = 0; r < 8; ++r) partial[wave * 16 + 8 + r] = acc[r];
    }
    __syncthreads();

    if (threadIdx.x < 16) {
        float s = fc_b[0];
#pragma unroll
        for (int w = 0; w < WAVES_PER_WG; ++w) s += partial[w * 16 + threadIdx.x];
        out[b * NN + n0 + threadIdx.x] = s;
    }
}

// ---------------------------------------------------------------------------
extern "C" void kernel_launch(void* const* d_in, const int* in_sizes, int n_in,
                              void* d_out, int out_size, void* d_ws, size_t ws_size,
                              hipStream_t stream) {
    const float* W    = (const float*)d_in[0];   // [BS, NN, NN, JJ]
    const float* x    = (const float*)d_in[1];   // [BS, NN, F_IN]
    const float* fc_w = (const float*)d_in[2];   // [1, JJ*F_IN]
    const float* fc_b = (const float*)d_in[3];   // [1]
    float*       out  = (float*)d_out;           // [BS, NN]
    float*       y    = (float*)d_ws;            // [BS, NN, JJ] = 96 KB scratch

    const int total_y = BS * NN * JJ;
    gconv_fold_fc<<<(total_y + 255) / 256, 256, 0, stream>>>(x, fc_w, y);

    gconv_gemv_wmma<<<(BS * NN) / 16, THREADS, 0, stream>>>(W, y, fc_b, out);
}